// ScaleHierarchicalOptimizer_37005438222751
// MI455X (gfx1250) — compile-verified
//
#include <hip/hip_runtime.h>
#include <math.h>

// Problem constants (from the reference)
#define A_DIM 1024
#define B_DIM 4096
#define NELEM (A_DIM * B_DIM)   // 4,194,304
#define T_DIM 4
#define PU 10
#define TU 5
#define GU 5

typedef float v2f __attribute__((ext_vector_type(2)));
typedef float v4f __attribute__((ext_vector_type(4)));
typedef float v8f __attribute__((ext_vector_type(8)));

__device__ __forceinline__ float sigm(float x) { return 1.0f / (1.0f + __expf(-x)); }

// ---------------------------------------------------------------------------
// k0: zero the reduction accumulators in workspace
// red[0]=sum(eta), red[1]=sum(d_theta^2), red[2..11]=colsum(param_state_new)
// ---------------------------------------------------------------------------
__global__ void k0_zero(float* __restrict__ red) {
    if (threadIdx.x < 12) red[threadIdx.x] = 0.0f;
}

// ---------------------------------------------------------------------------
// k1: elementwise pass 1 — EMA updates + partial reductions
// param_state staged through LDS with coalesced b128 loads.
// All big streams are non-temporal (single-use vs 192MB L2).
// ---------------------------------------------------------------------------
__global__ __launch_bounds__(256)
void k1_pass1(const float* __restrict__ grads,
              const float* __restrict__ g_bar,
              const float* __restrict__ lam,
              const float* __restrict__ eta_bar,
              const float* __restrict__ ps,
              const float* __restrict__ gamma_p,
              const float* __restrict__ W_dt, const float* __restrict__ b_dt,
              const float* __restrict__ W_dn, const float* __restrict__ b_dn,
              const float* __restrict__ W_bg, const float* __restrict__ b_bg,
              const float* __restrict__ W_bl, const float* __restrict__ b_bl,
              float* __restrict__ out_gbn,
              float* __restrict__ out_lmn,
              float* __restrict__ out_ebn,
              float* __restrict__ red)
{
    __shared__ float sps[256 * PU];

    const int tid = threadIdx.x;
    const size_t i = (size_t)blockIdx.x * 256 + tid;

    // coalesced b128 staging of this block's param_state slice (2560 floats)
    {
        const v4f* src = (const v4f*)(ps + (size_t)blockIdx.x * (256 * PU));
        v4f* dst = (v4f*)sps;
#pragma unroll
        for (int idx = tid; idx < (256 * PU) / 4; idx += 256)
            dst[idx] = __builtin_nontemporal_load(&src[idx]);
    }
    __syncthreads();

    float s[PU];
#pragma unroll
    for (int u = 0; u < PU; ++u) s[u] = sps[tid * PU + u];

    float dbg = b_bg[0], dbl = b_bl[0], ddn = b_dn[0], ddt = b_dt[0];
#pragma unroll
    for (int u = 0; u < PU; ++u) {
        dbg += s[u] * W_bg[u];
        dbl += s[u] * W_bl[u];
        ddn += s[u] * W_dn[u];
        ddt += s[u] * W_dt[u];
    }
    const float beta_g = sigm(dbg);
    const float beta_l = sigm(dbl);
    const float eta_pre = sigm(ddn);
    const float d_theta = ddt;

    const float g  = __builtin_nontemporal_load(&grads[i]);
    const float g2 = g * g;
    const float eb = __builtin_nontemporal_load(&eta_bar[i]);
    const float gamma = gamma_p[0];

    // b^(0.5^t): t=0 -> b, then sqrt chain
    float b1 = beta_g, b2 = beta_l;
#pragma unroll
    for (int t = 0; t < T_DIM; ++t) {
        const size_t off = (size_t)t * NELEM + i;
        const float gb = __builtin_nontemporal_load(&g_bar[off]);
        const float lm = __builtin_nontemporal_load(&lam[off]);
        __builtin_nontemporal_store(b1 * gb + (1.0f - b1) * g,  &out_gbn[off]);
        __builtin_nontemporal_store(b2 * lm + (1.0f - b2) * g2, &out_lmn[off]);
        b1 = sqrtf(b1);
        b2 = sqrtf(b2);
    }

    const float eta = eta_pre + eb;
    __builtin_nontemporal_store(gamma * eb + (1.0f - gamma) * eta, &out_ebn[i]);

    // wave32 reduction of eta and d_theta^2, then one atomic per wave
    float se = eta;
    float sd = d_theta * d_theta;
#pragma unroll
    for (int m = 16; m > 0; m >>= 1) {
        se += __shfl_xor(se, m, 32);
        sd += __shfl_xor(sd, m, 32);
    }
    if ((tid & 31) == 0) {
        atomicAdd(&red[0], se);
        atomicAdd(&red[1], sd);
    }
}

// ---------------------------------------------------------------------------
// k2: WMMA pass — GRU GEMMs via V_WMMA_F32_16X16X4_F32, delta_theta,
//     param_state_new, column-sum partials.
// param_state staged in via b128 (aliasing xg); param_state_new staged out
// via b128 (aliasing hg). Barriers order the buffer reuse.
// ---------------------------------------------------------------------------
__global__ __launch_bounds__(128)
void k2_pass2(const float* __restrict__ gbn,          // g_bar_new (in d_out)
              const float* __restrict__ lmn,          // lam_new   (in d_out)
              const float* __restrict__ eta_bar,
              const float* __restrict__ ps,
              const float* __restrict__ tensor_state,
              const float* __restrict__ global_state,
              const float* __restrict__ W_dt, const float* __restrict__ b_dt,
              const float* __restrict__ W_dn, const float* __restrict__ b_dn,
              const float* __restrict__ pk,           // 19 x 30
              const float* __restrict__ prk,          // 10 x 30
              const float* __restrict__ pb,           // 2 x 30
              float* __restrict__ red,                // [0..1] read, [2..11] accum
              float* __restrict__ out_delta,
              float* __restrict__ out_psn)
{
    __shared__ float xbuf[128][20];   // x rows, K padded 19->20
    __shared__ float hbuf[128][12];   // h rows, K padded 10->12
    __shared__ float xg[128][32];     // x @ pk  (cols padded 30->32); also ps stage
    __shared__ float hg[128][32];     // h @ prk;                     also psn stage
    __shared__ float Bx[20][32];      // pk padded
    __shared__ float Bh[12][32];      // prk padded
    __shared__ float pbs[64];         // pb[0] at [0..29], pb[1] at [32..61]

    const int tid = threadIdx.x;
    const size_t i = (size_t)blockIdx.x * 128 + tid;

    // ---- stage padded weight matrices ----
    for (int idx = tid; idx < 20 * 32; idx += 128) {
        const int r = idx >> 5, c = idx & 31;
        Bx[r][c] = (r < 19 && c < 30) ? pk[r * 30 + c] : 0.0f;
    }
    for (int idx = tid; idx < 12 * 32; idx += 128) {
        const int r = idx >> 5, c = idx & 31;
        Bh[r][c] = (r < 10 && c < 30) ? prk[r * 30 + c] : 0.0f;
    }
    if (tid < 64) {
        const int h = tid >> 5, c = tid & 31;
        pbs[tid] = (c < 30) ? pb[h * 30 + c] : 0.0f;
    }

    // ---- coalesced b128 staging of param_state (1280 floats), alias xg ----
    float* spst = &xg[0][0];
    {
        const v4f* src = (const v4f*)(ps + (size_t)blockIdx.x * (128 * PU));
        v4f* dst = (v4f*)spst;
#pragma unroll
        for (int idx = tid; idx < (128 * PU) / 4; idx += 128)
            dst[idx] = __builtin_nontemporal_load(&src[idx]);
    }
    __syncthreads();

    // ---- per-element feature build ----
    float s[PU];
#pragma unroll
    for (int u = 0; u < PU; ++u) {
        s[u] = spst[tid * PU + u];
        hbuf[tid][u] = s[u];
    }
    hbuf[tid][10] = 0.0f;
    hbuf[tid][11] = 0.0f;

    float ddn = b_dn[0], ddt = b_dt[0];
#pragma unroll
    for (int u = 0; u < PU; ++u) {
        ddn += s[u] * W_dn[u];
        ddt += s[u] * W_dt[u];
    }
    const float eta = sigm(ddn) + eta_bar[i];

    const float inv_n    = 1.0f / (float)NELEM;
    const float mean_eta = red[0] * inv_n;
    const float scale    = (float)NELEM * rsqrtf(red[1]);   // N / ||d_theta||
    __builtin_nontemporal_store(__expf(eta) * ddt * scale, &out_delta[i]);

    float ll[T_DIM];
    float mll = 0.0f;
#pragma unroll
    for (int t = 0; t < T_DIM; ++t) {
        const size_t off = (size_t)t * NELEM + i;
        const float gb = __builtin_nontemporal_load(&gbn[off]);
        const float lm = __builtin_nontemporal_load(&lmn[off]);
        xbuf[tid][t] = gb * rsqrtf(lm);       // m
        ll[t] = __logf(lm);
        mll += ll[t];
    }
    mll *= 0.25f;
#pragma unroll
    for (int t = 0; t < T_DIM; ++t) xbuf[tid][4 + t] = ll[t] - mll;  // gam_feat
    xbuf[tid][8] = eta - mean_eta;                                   // eta_rel
#pragma unroll
    for (int k = 0; k < TU; ++k) xbuf[tid][9 + k]  = tensor_state[k];
#pragma unroll
    for (int k = 0; k < GU; ++k) xbuf[tid][14 + k] = global_state[k];
    xbuf[tid][19] = 0.0f;                                            // K pad

    __syncthreads();   // also orders: spst reads done before xg overwritten

    // ---- WMMA: D = A(16x4) x B(4x16) + C, f32 ----
    // A layout: lane l: M = l&15, K = {2h, 2h+1}, h = l>>4
    // C layout: VGPR v, lane l -> row (v + 8h), col (l&15)
    const int lane = tid & 31;
    const int wave = tid >> 5;          // 0..3
    const int half = lane >> 4;         // 0/1
    const int l15  = lane & 15;

#pragma unroll
    for (int sub = 0; sub < 2; ++sub) {
        const int r0 = wave * 32 + sub * 16;
#pragma unroll
        for (int nt = 0; nt < 2; ++nt) {
            // xg tile: K = 20 -> 5 steps
            v8f acc = {};
#pragma unroll
            for (int ks = 0; ks < 5; ++ks) {
                const int k0 = ks * 4 + half * 2;
                v2f a, b;
                a.x = xbuf[r0 + l15][k0];
                a.y = xbuf[r0 + l15][k0 + 1];
                b.x = Bx[k0][nt * 16 + l15];
                b.y = Bx[k0 + 1][nt * 16 + l15];
                acc = __builtin_amdgcn_wmma_f32_16x16x4_f32(
                        false, a, false, b, (short)0, acc, false, false);
            }
#pragma unroll
            for (int v = 0; v < 8; ++v)
                xg[r0 + v + half * 8][nt * 16 + l15] = acc[v];

            // hg tile: K = 12 -> 3 steps
            v8f acc2 = {};
#pragma unroll
            for (int ks = 0; ks < 3; ++ks) {
                const int k0 = ks * 4 + half * 2;
                v2f a, b;
                a.x = hbuf[r0 + l15][k0];
                a.y = hbuf[r0 + l15][k0 + 1];
                b.x = Bh[k0][nt * 16 + l15];
                b.y = Bh[k0 + 1][nt * 16 + l15];
                acc2 = __builtin_amdgcn_wmma_f32_16x16x4_f32(
                        false, a, false, b, (short)0, acc2, false, false);
            }
#pragma unroll
            for (int v = 0; v < 8; ++v)
                hg[r0 + v + half * 8][nt * 16 + l15] = acc2[v];
        }
    }

    __syncthreads();

    // ---- finish GRU gates elementwise (all LDS reads -> registers) ----
    float hn[PU];
#pragma unroll
    for (int u = 0; u < PU; ++u) {
        const float xz = xg[tid][u]       + pbs[u];
        const float hz = hg[tid][u]       + pbs[32 + u];
        const float xr = xg[tid][10 + u]  + pbs[10 + u];
        const float hr = hg[tid][10 + u]  + pbs[32 + 10 + u];
        const float xh = xg[tid][20 + u]  + pbs[20 + u];
        const float hh = hg[tid][20 + u]  + pbs[32 + 20 + u];
        const float z = sigm(xz + hz);
        const float r = sigm(xr + hr);
        const float cand = tanhf(xh + r * hh);
        hn[u] = z * s[u] + (1.0f - z) * cand;
    }

    // ---- column-sum partials for mean(param_state_new, axis=0) ----
#pragma unroll
    for (int u = 0; u < PU; ++u) {
        float v = hn[u];
#pragma unroll
        for (int m = 16; m > 0; m >>= 1) v += __shfl_xor(v, m, 32);
        if (lane == 0) atomicAdd(&red[2 + u], v);
    }

    __syncthreads();   // hg reads done before overwrite

    // ---- stage param_state_new to LDS, then coalesced b128 NT store ----
    float* hflat = &hg[0][0];
#pragma unroll
    for (int u = 0; u < PU; ++u) hflat[tid * PU + u] = hn[u];

    __syncthreads();
    {
        v4f* dst = (v4f*)(out_psn + (size_t)blockIdx.x * (128 * PU));
        const v4f* src = (const v4f*)hflat;
#pragma unroll
        for (int idx = tid; idx < (128 * PU) / 4; idx += 128)
            __builtin_nontemporal_store(src[idx], &dst[idx]);
    }
}

// ---------------------------------------------------------------------------
// k3: tiny tensor-level GRU (15 -> 15, u = 5), single wave
// ---------------------------------------------------------------------------
__global__ void k3_tensor(const float* __restrict__ red,
                          const float* __restrict__ tensor_state,
                          const float* __restrict__ global_state,
                          const float* __restrict__ tk,   // 15 x 15
                          const float* __restrict__ trk,  // 5 x 15
                          const float* __restrict__ tb,   // 2 x 15
                          float* __restrict__ out_tsn)
{
    __shared__ float xg[15];
    __shared__ float hg[15];
    const int c = threadIdx.x;
    const float inv_n = 1.0f / (float)NELEM;
    if (c < 15) {
        float acc = tb[c];
#pragma unroll
        for (int k = 0; k < 15; ++k) {
            const float xk = (k < PU) ? red[2 + k] * inv_n : global_state[k - PU];
            acc += xk * tk[k * 15 + c];
        }
        xg[c] = acc;
        float acc2 = tb[15 + c];
#pragma unroll
        for (int k = 0; k < TU; ++k) acc2 += tensor_state[k] * trk[k * 15 + c];
        hg[c] = acc2;
    }
    __syncthreads();
    if (c < TU) {
        const float z = sigm(xg[c] + hg[c]);
        const float r = sigm(xg[5 + c] + hg[5 + c]);
        const float cand = tanhf(xg[10 + c] + r * hg[10 + c]);
        out_tsn[c] = z * tensor_state[c] + (1.0f - z) * cand;
    }
}

// ---------------------------------------------------------------------------
extern "C" void kernel_launch(void* const* d_in, const int* in_sizes, int n_in,
                              void* d_out, int out_size, void* d_ws, size_t ws_size,
                              hipStream_t stream)
{
    // input order matches setup_inputs() dict
    const float* grads        = (const float*)d_in[1];
    const float* g_bar        = (const float*)d_in[2];
    const float* lam          = (const float*)d_in[3];
    const float* eta_bar      = (const float*)d_in[4];
    const float* param_state  = (const float*)d_in[5];
    const float* tensor_state = (const float*)d_in[6];
    const float* global_state = (const float*)d_in[7];
    const float* gamma        = (const float*)d_in[8];
    const float* W_dt = (const float*)d_in[9];
    const float* b_dt = (const float*)d_in[10];
    const float* W_dn = (const float*)d_in[11];
    const float* b_dn = (const float*)d_in[12];
    const float* W_bg = (const float*)d_in[13];
    const float* b_bg = (const float*)d_in[14];
    const float* W_bl = (const float*)d_in[15];
    const float* b_bl = (const float*)d_in[16];
    const float* pk   = (const float*)d_in[17];
    const float* prk  = (const float*)d_in[18];
    const float* pb   = (const float*)d_in[19];
    const float* tk   = (const float*)d_in[20];
    const float* trk  = (const float*)d_in[21];
    const float* tb   = (const float*)d_in[22];

    float* out = (float*)d_out;
    float* out_delta = out;                              // N
    float* out_psn   = out + (size_t)NELEM;              // 10N
    float* out_tsn   = out + (size_t)11 * NELEM;         // 5
    float* out_gbn   = out + (size_t)11 * NELEM + 5;     // 4N
    float* out_lmn   = out + (size_t)15 * NELEM + 5;     // 4N
    float* out_ebn   = out + (size_t)19 * NELEM + 5;     // N

    float* red = (float*)d_ws;                           // 12 accumulators

    k0_zero<<<1, 32, 0, stream>>>(red);

    k1_pass1<<<NELEM / 256, 256, 0, stream>>>(
        grads, g_bar, lam, eta_bar, param_state, gamma,
        W_dt, b_dt, W_dn, b_dn, W_bg, b_bg, W_bl, b_bl,
        out_gbn, out_lmn, out_ebn, red);

    k2_pass2<<<NELEM / 128, 128, 0, stream>>>(
        out_gbn, out_lmn, eta_bar, param_state, tensor_state, global_state,
        W_dt, b_dt, W_dn, b_dn, pk, prk, pb, red,
        out_delta, out_psn);

    k3_tensor<<<1, 32, 0, stream>>>(
        red, tensor_state, global_state, tk, trk, tb, out_tsn);
}